// IAT_64690797412572
// MI455X (gfx1250) — compile-verified
//
#include <hip/hip_runtime.h>

#define HW_PIX (1024 * 1024)
#define NBINS 256
#define BATCH 16
#define BLKX 512   // blocks per image in kernel 2

typedef __attribute__((ext_vector_type(2))) float v2f;
typedef __attribute__((ext_vector_type(8))) float v8f;

typedef __attribute__((address_space(1))) int* gas_i32p;  // global AS pointer
typedef __attribute__((address_space(3))) int* las_i32p;  // LDS AS pointer

#if __has_builtin(__builtin_amdgcn_wmma_f32_16x16x4_f32)
#define HAS_WMMA_F32X4 1
#endif
#if __has_builtin(__builtin_amdgcn_global_load_async_to_lds_b32)
#define HAS_ASYNC_LDS 1
#endif

// pow(x, g) for x in [1e-8, 1]: v_log_f32 is log2, v_exp_f32 is exp2.
__device__ __forceinline__ float fast_pow_clamped(float x, float g) {
  return __builtin_amdgcn_exp2f(g * __builtin_amdgcn_logf(x));
}

// ---------------------------------------------------------------------------
// Kernel 1: build per-channel monotone curves: sigmoid -> cumsum -> global
// min/max normalize (min/max over the whole [B,256] array per channel).
// Output layout: curve[b][ch][bin]  (ch: 0=r,1=g,2=b)
// ---------------------------------------------------------------------------
__global__ __launch_bounds__(256) void iat_build_curves(
    const float* __restrict__ rr, const float* __restrict__ gr,
    const float* __restrict__ br, float* __restrict__ curve) {
  __shared__ float raw[BATCH][NBINS];
  __shared__ float sc[NBINS];
  __shared__ float s_mn, s_mx;
  const int t = threadIdx.x;

  for (int ch = 0; ch < 3; ++ch) {
    const float* src = (ch == 0) ? rr : (ch == 1) ? gr : br;
    if (t == 0) { s_mn = 3.4e38f; s_mx = -3.4e38f; }
    __syncthreads();
    for (int b = 0; b < BATCH; ++b) {
      float x = src[b * NBINS + t];
      sc[t] = 1.0f / (1.0f + __expf(-x));  // sigmoid
      __syncthreads();
      // Hillis-Steele inclusive scan over 256 bins
      for (int off = 1; off < NBINS; off <<= 1) {
        float y = (t >= off) ? sc[t - off] : 0.0f;
        __syncthreads();
        sc[t] += y;
        __syncthreads();
      }
      raw[b][t] = sc[t];
      // monotone rows: row-min = sc[0], row-max = sc[255]
      if (t == 0) s_mn = fminf(s_mn, sc[0]);
      if (t == NBINS - 1) s_mx = fmaxf(s_mx, sc[NBINS - 1]);
      __syncthreads();
    }
    const float mn = s_mn;
    const float inv = 1.0f / (s_mx - mn);
    for (int b = 0; b < BATCH; ++b)
      curve[(b * 3 + ch) * NBINS + t] = (raw[b][t] - mn) * inv;
    __syncthreads();
  }
}

// ---------------------------------------------------------------------------
// Kernel 2: LUT gather (LDS) + 3x3 color matrix via V_WMMA_F32_16X16X4_F32
// + clamp + per-image gamma. grid = (BLKX, BATCH), block = 256 (8 waves).
// Each wave processes 32 consecutive pixels per iteration (coalesced b32).
// ---------------------------------------------------------------------------
__global__ __launch_bounds__(256) void iat_apply(
    const float* __restrict__ img,     // [B,3,HW]
    const float* __restrict__ curve,   // [B,3,256]
    const float* __restrict__ color,   // [B,3,3]
    const float* __restrict__ gamma,   // [B]
    float* __restrict__ out) {         // [B,3,HW]
  extern __shared__ float lut[];  // 3*256 floats
  const int b = blockIdx.y;
  const int tid = threadIdx.x;
  const int lane = tid & 31;
  const int wid = tid >> 5;

  // ---- Stage this image's LUT into LDS (async global->LDS on CDNA5) ----
  const float* gsrc = curve + b * 3 * NBINS;
#ifdef HAS_ASYNC_LDS
  for (int i = tid; i < 3 * NBINS; i += 256) {
    __builtin_amdgcn_global_load_async_to_lds_b32(
        (gas_i32p)(gsrc + i), (las_i32p)(lut + i), 0, 0);
  }
#if __has_builtin(__builtin_amdgcn_s_wait_asynccnt)
  __builtin_amdgcn_s_wait_asynccnt(0);
#else
  asm volatile("s_wait_asynccnt 0" ::: "memory");
#endif
#else
  for (int i = tid; i < 3 * NBINS; i += 256) lut[i] = gsrc[i];
#endif
  __syncthreads();

  const float gma = gamma[b];
  const float* cm = color + b * 9;  // cm[d*3+c] ; out[d] = sum_c cm[d][c]*in[c]

#ifdef HAS_WMMA_F32X4
  // A (16x4 f32, 2 VGPRs): VGPR0 = K{0|2}, VGPR1 = K{1|3}; M = lane%16.
  v2f A;
  {
    const int m = lane & 15;
    const int k0 = (lane < 16) ? 0 : 2;
    const int k1 = (lane < 16) ? 1 : 3;
    A.x = (m < 3) ? cm[m * 3 + k0] : 0.0f;
    A.y = (m < 3 && k1 < 3) ? cm[m * 3 + k1] : 0.0f;
  }
#else
  const float c00 = cm[0], c01 = cm[1], c02 = cm[2];
  const float c10 = cm[3], c11 = cm[4], c12 = cm[5];
  const float c20 = cm[6], c21 = cm[7], c22 = cm[8];
#endif

  const int chunk = HW_PIX / BLKX;  // pixels per block
  const int base = blockIdx.x * chunk;
  const int end = base + chunk;
  const float* imgB = img + (size_t)b * 3 * HW_PIX;
  float* outB = out + (size_t)b * 3 * HW_PIX;

  for (int p0 = base + wid * 32; p0 < end; p0 += 8 * 32) {
    const int p = p0 + lane;
    const float r = imgB[p];
    const float g = imgB[HW_PIX + p];
    const float bl = imgB[2 * HW_PIX + p];

    if (p0 + 8 * 32 < end) {  // prefetch next iteration's streams
      __builtin_prefetch(imgB + p + 8 * 32, 0, 0);
      __builtin_prefetch(imgB + HW_PIX + p + 8 * 32, 0, 0);
      __builtin_prefetch(imgB + 2 * HW_PIX + p + 8 * 32, 0, 0);
    }

    // LUT gather from LDS (truncating index, as in reference int32 cast)
    const int ir = (int)fminf(fmaxf(r * 255.0f, 0.0f), 255.0f);
    const int ig = (int)fminf(fmaxf(g * 255.0f, 0.0f), 255.0f);
    const int ib = (int)fminf(fmaxf(bl * 255.0f, 0.0f), 255.0f);
    const float rh = lut[ir];
    const float gh = lut[NBINS + ig];
    const float bh = lut[2 * NBINS + ib];

    float o0, o1, o2;  // output channels for this lane's pixel
#ifdef HAS_WMMA_F32X4
    // B (4x16 f32, 2 VGPRs): VGPR0 = rows K{0|2}, VGPR1 = rows K{1|3}.
    // WMMA #1 covers pixels (lanes) 0..15, WMMA #2 covers 16..31.
    const float bh_lo = __shfl(bh, lane & 15, 32);         // bh of lanes 0..15
    const float rh_hi = __shfl(rh, (lane & 15) + 16, 32);  // rh of lanes 16..31
    const float gh_hi = __shfl(gh, (lane & 15) + 16, 32);  // gh of lanes 16..31
    v2f B1, B2;
    B1.x = (lane < 16) ? rh : bh_lo;  // K0 = r | K2 = b
    B1.y = (lane < 16) ? gh : 0.0f;   // K1 = g | K3 = 0
    B2.x = (lane < 16) ? rh_hi : bh;
    B2.y = (lane < 16) ? gh_hi : 0.0f;
    v8f Z = {};
    v8f D1 = __builtin_amdgcn_wmma_f32_16x16x4_f32(
        false, A, false, B1, (short)0, Z, false, false);
    v8f D2 = __builtin_amdgcn_wmma_f32_16x16x4_f32(
        false, A, false, B2, (short)0, Z, false, false);
    // D vgpr c holds out-channel c for pixels 0..15 in lanes 0..15.
    const float h0 = __shfl(D2[0], lane & 15, 32);
    const float h1 = __shfl(D2[1], lane & 15, 32);
    const float h2 = __shfl(D2[2], lane & 15, 32);
    o0 = (lane < 16) ? D1[0] : h0;
    o1 = (lane < 16) ? D1[1] : h1;
    o2 = (lane < 16) ? D1[2] : h2;
#else
    o0 = c00 * rh + c01 * gh + c02 * bh;
    o1 = c10 * rh + c11 * gh + c12 * bh;
    o2 = c20 * rh + c21 * gh + c22 * bh;
#endif
    o0 = fast_pow_clamped(fminf(fmaxf(o0, 1e-8f), 1.0f), gma);
    o1 = fast_pow_clamped(fminf(fmaxf(o1, 1e-8f), 1.0f), gma);
    o2 = fast_pow_clamped(fminf(fmaxf(o2, 1e-8f), 1.0f), gma);
    outB[p] = o0;
    outB[HW_PIX + p] = o1;
    outB[2 * HW_PIX + p] = o2;
  }
}

// ---------------------------------------------------------------------------
extern "C" void kernel_launch(void* const* d_in, const int* in_sizes, int n_in,
                              void* d_out, int out_size, void* d_ws,
                              size_t ws_size, hipStream_t stream) {
  const float* img = (const float*)d_in[0];    // [16,3,1024,1024]
  const float* rr = (const float*)d_in[1];     // [16,256]
  const float* gr = (const float*)d_in[2];     // [16,256]
  const float* br = (const float*)d_in[3];     // [16,256]
  const float* gamma = (const float*)d_in[4];  // [16,1]
  const float* color = (const float*)d_in[5];  // [16,3,3]
  float* out = (float*)d_out;
  float* curve = (float*)d_ws;  // 16*3*256 floats = 48 KB scratch

  iat_build_curves<<<1, 256, 0, stream>>>(rr, gr, br, curve);

  dim3 grid(BLKX, BATCH);
  iat_apply<<<grid, 256, 3 * NBINS * sizeof(float), stream>>>(
      img, curve, color, gamma, out);
}